// EntityResolution_4939212390964
// MI455X (gfx1250) — compile-verified
//
#include <hip/hip_runtime.h>
#include <math.h>

#define BB 8
#define SS 16
#define CC 32
#define PP 64
#define EE 768
#define MM 512      // S*C
#define LL 32768    // M*P
#define NEc 1000000

typedef __attribute__((ext_vector_type(2))) float v2f;
typedef __attribute__((ext_vector_type(8))) float v8f;

// ---------------- Kernel 1: embedding bag (pure bandwidth) ----------------
// One 256-thread block per bag (b,m). Each thread accumulates 3 floats of the
// 768-wide row sum across the 64 gathered rows. Reads are 3 coalesced 1KB
// segments per row.
__global__ void er_bag_kernel(const int* __restrict__ ids,
                              const float* __restrict__ att,
                              const float* __restrict__ table,
                              float* __restrict__ kb) {
  int bag = blockIdx.x;            // b*MM + m
  int b = bag >> 9;
  int m = bag & 511;
  __shared__ int   sid[PP];
  __shared__ float sw[PP];
  int t = threadIdx.x;
  if (t < PP) {
    int l = b * LL + m * PP + t;
    sid[t] = ids[l];
    sw[t]  = att[l];
  }
  __syncthreads();
  float a0 = 0.f, a1 = 0.f, a2 = 0.f;
  int e0 = t, e1 = t + 256, e2 = t + 512;
  for (int p = 0; p < PP; ++p) {
    const float* row = table + (size_t)sid[p] * EE;
    float w = sw[p];
    a0 = fmaf(row[e0], w, a0);
    a1 = fmaf(row[e1], w, a1);
    a2 = fmaf(row[e2], w, a2);
  }
  float* dst = kb + (size_t)bag * EE;
  dst[e0] = a0; dst[e1] = a1; dst[e2] = a2;
}

// ---------------- Kernel 2: WMMA GEMM 512x768 x 768x16 per batch ----------
// One wave (32 threads) per 16-row tile of kb. C accumulates in 8 VGPRs.
// A layout (32-bit 16x4): lane -> M=lane%16, VGPR v -> K = 2*(lane/16)+v
// B layout (4x16):        lane -> N=lane%16, VGPR v -> K = 2*(lane/16)+v
// D layout:               VGPR r -> M = r + 8*(lane/16), N = lane%16
__global__ void er_wmma_gemm_kernel(const float* __restrict__ kb,
                                    const float* __restrict__ span,
                                    float* __restrict__ s1) {
  int blk  = blockIdx.x;        // b*32 + tile
  int b    = blk >> 5;
  int tile = blk & 31;
  int lane = threadIdx.x;       // 0..31
  int n    = lane & 15;
  int kh   = lane >> 4;         // 0 or 1
  int row  = tile * 16 + n;
  const float* arow = kb   + ((size_t)b * MM + row) * EE;
  const float* brow = span + ((size_t)b * SS + n)   * EE;
  v8f c = {};
  for (int k0 = 0; k0 < EE; k0 += 4) {
    float2 av = *(const float2*)(arow + k0 + 2 * kh);
    float2 bv = *(const float2*)(brow + k0 + 2 * kh);
    v2f a, bf;
    a[0] = av.x; a[1] = av.y;
    bf[0] = bv.x; bf[1] = bv.y;
    c = __builtin_amdgcn_wmma_f32_16x16x4_f32(false, a, false, bf,
                                              (short)0, c, false, false);
  }
  float* out = s1 + ((size_t)b * MM + tile * 16) * 16;
#pragma unroll
  for (int r = 0; r < 8; ++r)
    out[(r + 8 * kh) * 16 + n] = c[r];
}

// ---------------- Kernel 3: softmaxes + sparse scatter stats --------------
// One 512-thread block per batch.
__global__ void er_score_kernel(const float* __restrict__ s1,
                                const float* __restrict__ span,
                                const float* __restrict__ spanW,
                                const float* __restrict__ spanb,
                                const int*  __restrict__ qids,
                                float* __restrict__ base,
                                int*   __restrict__ fix_qid,
                                float* __restrict__ fix_val) {
  int b = blockIdx.x;
  int t = threadIdx.x;                 // 0..511
  __shared__ float sscore[SS];
  __shared__ float shv[512];
  __shared__ float buf[512];
  __shared__ int   qsh[512];
  __shared__ float cands[512];

  int wave = t >> 5, lane = t & 31;
  // span_scores[s] = span[b,s,:] . spanW + spanb  (wave `wave` computes s=wave)
  {
    const float* sp = span + ((size_t)b * SS + wave) * EE;
    float acc = 0.f;
    for (int e = lane; e < EE; e += 32) acc = fmaf(sp[e], spanW[e], acc);
    for (int off = 16; off > 0; off >>= 1) acc += __shfl_xor(acc, off, 32);
    if (lane == 0) sscore[wave] = acc + spanb[0];
  }
  __syncthreads();

  int m = t, s = t >> 5, c = t & 31, k = c & 15;
  float x = s1[((size_t)b * MM + m) * 16 + k];   // sum1[b,s,c]
  shv[t] = x;
  __syncthreads();
  // softmax over s (axis=1) for fixed c
  float mx = -3.402823466e38f;
  for (int u = 0; u < SS; ++u) mx = fmaxf(mx, shv[u * 32 + c]);
  float den = 0.f;
  for (int u = 0; u < SS; ++u) den += expf(shv[u * 32 + c] - mx);
  float y = sscore[s] * (expf(x - mx) / den);    // mult2[b, m]
  __syncthreads();

  // softmax over all 512 (cand_scores)
  buf[t] = y; __syncthreads();
  for (int off = 256; off > 0; off >>= 1) {
    if (t < off) buf[t] = fmaxf(buf[t], buf[t + off]);
    __syncthreads();
  }
  float gmx = buf[0]; __syncthreads();
  float ex = expf(y - gmx);
  buf[t] = ex; __syncthreads();
  for (int off = 256; off > 0; off >>= 1) {
    if (t < off) buf[t] += buf[t + off];
    __syncthreads();
  }
  float gsum = buf[0]; __syncthreads();
  float cand = ex / gsum;
  cands[t] = cand;
  int qid = qids[(size_t)b * MM + t];
  qsh[t] = qid;
  __syncthreads();

  // deduplicate scatter indices: first occurrence owns the summed value
  bool first = (qid < NEc);
  if (first) {
    for (int u = 0; u < t; ++u)
      if (qsh[u] == qid) { first = false; break; }
  }
  float vsum = 0.f;
  if (first) {
    for (int u = 0; u < 512; ++u)
      if (qsh[u] == qid) vsum += cands[u];
  }

  // per-batch max over x0 (zeros elsewhere => clamp at 0)
  buf[t] = first ? vsum : -3.402823466e38f; __syncthreads();
  for (int off = 256; off > 0; off >>= 1) {
    if (t < off) buf[t] = fmaxf(buf[t], buf[t + off]);
    __syncthreads();
  }
  float bmx = fmaxf(buf[0], 0.f); __syncthreads();

  buf[t] = first ? 1.f : 0.f; __syncthreads();
  for (int off = 256; off > 0; off >>= 1) {
    if (t < off) buf[t] += buf[t + off];
    __syncthreads();
  }
  float cnt = buf[0]; __syncthreads();

  buf[t] = first ? expf(vsum - bmx) : 0.f; __syncthreads();
  for (int off = 256; off > 0; off >>= 1) {
    if (t < off) buf[t] += buf[t + off];
    __syncthreads();
  }
  float Z = buf[0] + ((float)NEc - cnt) * expf(-bmx);

  if (t == 0) base[b] = expf(-bmx) / Z;
  fix_qid[b * MM + t] = first ? qid : -1;
  fix_val[b * MM + t] = first ? (expf(vsum - bmx) / Z) : 0.f;
}

// ---------------- Kernel 4: dense fill with default softmax value ---------
__global__ void er_fill_kernel(float* __restrict__ out,
                               const float* __restrict__ base) {
  int b = blockIdx.y;
  int i = blockIdx.x * blockDim.x + threadIdx.x;
  if (i < NEc) out[(size_t)b * NEc + i] = base[b];
}

// ---------------- Kernel 5: overwrite scattered positions -----------------
__global__ void er_fixup_kernel(float* __restrict__ out,
                                const int* __restrict__ fix_qid,
                                const float* __restrict__ fix_val) {
  int b = blockIdx.x;
  int t = threadIdx.x;
  int q = fix_qid[b * MM + t];
  if (q >= 0) out[(size_t)b * NEc + q] = fix_val[b * MM + t];
}

extern "C" void kernel_launch(void* const* d_in, const int* in_sizes, int n_in,
                              void* d_out, int out_size, void* d_ws, size_t ws_size,
                              hipStream_t stream) {
  const float* span  = (const float*)d_in[0];   // (B,S,E)
  const int*   ids   = (const int*)  d_in[1];   // (B,L)
  // d_in[2] = offsets_tr: fixed arange(M)*P, not needed
  const float* att   = (const float*)d_in[3];   // (B,L)
  const int*   qids  = (const int*)  d_in[4];   // (B,M)
  const float* table = (const float*)d_in[5];   // (T,E)
  const float* spanW = (const float*)d_in[6];   // (E,1)
  const float* spanb = (const float*)d_in[7];   // (1,)
  float* out = (float*)d_out;                   // (B,NE)

  // workspace layout (256B aligned chunks)
  char* ws = (char*)d_ws;
  size_t off = 0;
  float* kb = (float*)(ws + off);  off += (size_t)BB * MM * EE * sizeof(float);  // 12.6 MB
  off = (off + 255) & ~(size_t)255;
  float* s1 = (float*)(ws + off);  off += (size_t)BB * MM * 16 * sizeof(float);  // 256 KB
  off = (off + 255) & ~(size_t)255;
  float* base = (float*)(ws + off); off += BB * sizeof(float);
  off = (off + 255) & ~(size_t)255;
  int* fix_qid = (int*)(ws + off);  off += (size_t)BB * MM * sizeof(int);
  off = (off + 255) & ~(size_t)255;
  float* fix_val = (float*)(ws + off); off += (size_t)BB * MM * sizeof(float);

  er_bag_kernel<<<BB * MM, 256, 0, stream>>>(ids, att, table, kb);
  er_wmma_gemm_kernel<<<BB * 32, 32, 0, stream>>>(kb, span, s1);
  er_score_kernel<<<BB, 512, 0, stream>>>(s1, span, spanW, spanb, qids,
                                          base, fix_qid, fix_val);
  dim3 fg((NEc + 255) / 256, BB);
  er_fill_kernel<<<fg, 256, 0, stream>>>(out, base);
  er_fixup_kernel<<<BB, MM, 0, stream>>>(out, fix_qid, fix_val);
}